// DescriptorLoss_67095979098625
// MI455X (gfx1250) — compile-verified
//
#include <hip/hip_runtime.h>
#include <hip/hip_bf16.h>

typedef float v2f __attribute__((ext_vector_type(2)));
typedef float v8f __attribute__((ext_vector_type(8)));

namespace {
constexpr int B_    = 2;
constexpr int DDIM  = 256;
constexpr int HC    = 60;
constexpr int WC    = 80;
constexpr int NPOS  = HC * WC;     // 4800
constexpr int NT    = NPOS / 16;   // 300 16-wide tiles
constexpr int NPAIR = NT / 2;      // 150 tile-pairs (exact)
constexpr int ASTR  = 260;         // padded LDS row stride (floats) -> conflict-free b64 reads

// workspace layout (float offsets)
constexpr size_t SZ_MAT     = (size_t)B_ * NPOS * DDIM;     // 2,457,600
constexpr size_t OFF_DESCN  = 0;
constexpr size_t OFF_WARPN  = OFF_DESCN  + SZ_MAT;
constexpr size_t OFF_WCOORD = OFF_WARPN  + SZ_MAT;          // B*N*2
constexpr size_t OFF_VALID  = OFF_WCOORD + (size_t)B_*NPOS*2;
constexpr size_t OFF_RSQ    = OFF_VALID  + (size_t)B_*NPOS;
constexpr size_t OFF_INVR   = OFF_RSQ    + (size_t)B_*NPOS;
constexpr size_t OFF_CSQ    = OFF_INVR   + (size_t)B_*NPOS;
constexpr size_t OFF_INVC   = OFF_CSQ    + (size_t)B_*NPOS;
constexpr size_t OFF_BSUM   = OFF_INVC   + (size_t)B_*NPOS;
constexpr size_t OFF_INORM  = OFF_BSUM   + B_;
} // namespace

// ---------------------------------------------------------------- prep kernels

__global__ void zero_kernel(float* __restrict__ out, float* __restrict__ bsum) {
  if (threadIdx.x == 0) out[0] = 0.f;
  if (threadIdx.x < B_) bsum[threadIdx.x] = 0.f;
}

// one block per row; l2-normalize 256-float rows of DESC / warp_DESC
__global__ __launch_bounds__(256) void row_l2norm_kernel(
    const float* __restrict__ DESC, const float* __restrict__ WDESC,
    float* __restrict__ descN, float* __restrict__ warpN) {
  int idx = blockIdx.x;
  const int nrows = B_ * NPOS;
  const float* src; float* dst;
  if (idx < nrows) { src = DESC  + (size_t)idx * DDIM; dst = descN + (size_t)idx * DDIM; }
  else { idx -= nrows; src = WDESC + (size_t)idx * DDIM; dst = warpN + (size_t)idx * DDIM; }
  float x = src[threadIdx.x];
  float s = x * x;
  #pragma unroll
  for (int m = 16; m >= 1; m >>= 1) s += __shfl_xor(s, m, 32);
  __shared__ float ws[8];
  __shared__ float invn;
  int lane = threadIdx.x & 31, wid = threadIdx.x >> 5;
  if (lane == 0) ws[wid] = s;
  __syncthreads();
  if (threadIdx.x == 0) {
    float t = 0.f;
    #pragma unroll
    for (int i = 0; i < 8; i++) t += ws[i];
    invn = 1.0f / fmaxf(sqrtf(t), 1e-12f);
  }
  __syncthreads();
  dst[threadIdx.x] = x * invn;
}

// warp coords per source cell p: homography, z-clamp, store (wx, wy)
__global__ void warp_coords_kernel(const float* __restrict__ Hinv, float* __restrict__ wc) {
  int idx = blockIdx.x * blockDim.x + threadIdx.x;
  if (idx >= B_ * NPOS) return;
  int b = idx / NPOS, p = idx % NPOS;
  int i = p / WC, j = p % WC;
  float cx = j * 8.f + 4.f, cy = i * 8.f + 4.f;
  const float* h = Hinv + b * 9;
  float qx = h[0]*cx + h[1]*cy + h[2];
  float qy = h[3]*cx + h[4]*cy + h[5];
  float qz = h[6]*cx + h[7]*cy + h[8];
  float z  = (fabsf(qz) > 1e-8f) ? qz : 1e-8f;
  wc[(size_t)idx*2 + 0] = qx / z;
  wc[(size_t)idx*2 + 1] = qy / z;
}

// valid[b,q] = product of the 8x8 v_mask block; accumulate per-batch sums
__global__ void valid_kernel(const float* __restrict__ vmask,
                             float* __restrict__ valid, float* __restrict__ bsum) {
  int idx = blockIdx.x * blockDim.x + threadIdx.x;
  if (idx >= B_ * NPOS) return;
  int b = idx / NPOS, p = idx % NPOS;
  int h = p / WC, w = p % WC;
  const float* m = vmask + (size_t)b * (HC*8) * (WC*8) + (size_t)(h*8) * (WC*8) + w*8;
  float prod = 1.f;
  #pragma unroll
  for (int dy = 0; dy < 8; dy++)
    #pragma unroll
    for (int dx = 0; dx < 8; dx++) prod *= m[dy * (WC*8) + dx];
  valid[idx] = prod;
  atomicAdd(&bsum[b], prod);
}

__global__ void finalize_norm_kernel(const float* __restrict__ bsum, float* __restrict__ inorm) {
  if (threadIdx.x < B_) inorm[threadIdx.x] = 1.0f / (bsum[threadIdx.x] * (float)NPOS);
}

__global__ void inv_sqrt_kernel(const float* __restrict__ in, float* __restrict__ out, int n) {
  int idx = blockIdx.x * blockDim.x + threadIdx.x;
  if (idx < n) out[idx] = 1.0f / fmaxf(sqrtf(in[idx]), 1e-12f);
}

// ------------------------------------------------- WMMA strip sum-of-squares
// One workgroup owns 16 rows of the (M=Amat rows, N=Bmat rows) Gram matrix.
// Each wave processes a pair of 16-column tiles with two independent WMMA
// accumulator chains (shared A fragment), out[row] = sum_n relu(g)^2 * scale[n]^2.
__global__ __launch_bounds__(256) void strip_sumsq_kernel(
    const float* __restrict__ Amat, const float* __restrict__ Bmat,
    const float* __restrict__ scale, float* __restrict__ out) {
  __shared__ float As[16 * ASTR];
  __shared__ float red[16];
  int b = blockIdx.x / NT, rt = blockIdx.x % NT;
  int rowbase = rt * 16;
  const float* Arow = Amat + ((size_t)b * NPOS + rowbase) * DDIM;
  {
    int t = threadIdx.x, r = t >> 4, c0 = (t & 15) * 16;
    #pragma unroll
    for (int i = 0; i < 4; i++)
      *(float4*)&As[r * ASTR + c0 + i*4] = *(const float4*)(Arow + r * DDIM + c0 + i*4);
  }
  if (threadIdx.x < 16) red[threadIdx.x] = 0.f;
  __syncthreads();

  int lane = threadIdx.x & 31, wid = threadIdx.x >> 5;
  int half = lane >> 4, r16 = lane & 15;
  const bool has_scale = (scale != nullptr);
  const float* Alds  = &As[r16 * ASTR + 2*half];
  const float* Bbase = Bmat + (size_t)b * NPOS * DDIM;
  const float* sbase = scale ? (scale + (size_t)b * NPOS) : nullptr;

  float acc[8] = {0.f,0.f,0.f,0.f,0.f,0.f,0.f,0.f};
  for (int t = wid; t < NPAIR; t += 8) {      // tiles 2t and 2t+1
    int n0 = t * 32 + r16;
    const float* Brow0 = Bbase + (size_t)n0 * DDIM + 2*half;
    const float* Brow1 = Brow0 + (size_t)16 * DDIM;
    v8f c0 = {}, c1 = {};
    #pragma unroll 4
    for (int kk = 0; kk < DDIM; kk += 4) {
      v2f a  = *(const v2f*)(Alds  + kk);
      v2f b0 = *(const v2f*)(Brow0 + kk);
      v2f b1 = *(const v2f*)(Brow1 + kk);
      c0 = __builtin_amdgcn_wmma_f32_16x16x4_f32(false, a, false, b0, (short)0, c0, false, false);
      c1 = __builtin_amdgcn_wmma_f32_16x16x4_f32(false, a, false, b1, (short)0, c1, false, false);
    }
    float sc0 = 1.0f, sc1 = 1.0f;
    if (has_scale) {
      float s0 = sbase[n0];      sc0 = s0 * s0;
      float s1 = sbase[n0 + 16]; sc1 = s1 * s1;
    }
    #pragma unroll
    for (int v = 0; v < 8; v++) {
      float t0 = fmaxf(c0[v], 0.f);
      float t1 = fmaxf(c1[v], 0.f);
      acc[v] += t0 * t0 * sc0 + t1 * t1 * sc1;
    }
  }
  // sum over N: reduce across the 16 lanes of each half
  #pragma unroll
  for (int v = 0; v < 8; v++) {
    float s = acc[v];
    s += __shfl_xor(s, 1, 16);
    s += __shfl_xor(s, 2, 16);
    s += __shfl_xor(s, 4, 16);
    s += __shfl_xor(s, 8, 16);
    acc[v] = s;
  }
  if (r16 == 0) {
    #pragma unroll
    for (int v = 0; v < 8; v++) atomicAdd(&red[v + 8*half], acc[v]);  // ds_add_f32
  }
  __syncthreads();
  if (threadIdx.x < 16) out[(size_t)b * NPOS + rowbase + threadIdx.x] = red[threadIdx.x];
}

// ---------------------------------------------------------- fused loss kernel
__global__ __launch_bounds__(256) void loss_kernel(
    const float* __restrict__ descN, const float* __restrict__ warpN,
    const float* __restrict__ inv_r, const float* __restrict__ inv_c,
    const float* __restrict__ wc, const float* __restrict__ valid,
    const float* __restrict__ inorm, float* __restrict__ out) {
  __shared__ float As[16 * ASTR];
  __shared__ float red;
  int b = blockIdx.x / NT, rt = blockIdx.x % NT;
  int rowbase = rt * 16;
  const float* Arow = descN + ((size_t)b * NPOS + rowbase) * DDIM;
  {
    int t = threadIdx.x, r = t >> 4, c0 = (t & 15) * 16;
    #pragma unroll
    for (int i = 0; i < 4; i++)
      *(float4*)&As[r * ASTR + c0 + i*4] = *(const float4*)(Arow + r * DDIM + c0 + i*4);
  }
  if (threadIdx.x == 0) red = 0.f;
  __syncthreads();

  int lane = threadIdx.x & 31, wid = threadIdx.x >> 5;
  int half = lane >> 4, r16 = lane & 15;
  float irow[8], wx[8], wy[8];
  #pragma unroll
  for (int v = 0; v < 8; v++) {
    int p = rowbase + v + 8*half;          // matches WMMA C-layout M index
    irow[v] = inv_r[(size_t)b * NPOS + p];
    wx[v]   = wc[((size_t)b * NPOS + p)*2 + 0];
    wy[v]   = wc[((size_t)b * NPOS + p)*2 + 1];
  }
  const float* Alds  = &As[r16 * ASTR + 2*half];
  const float* Bbase = warpN + (size_t)b * NPOS * DDIM;
  const float* icb   = inv_c + (size_t)b * NPOS;
  const float* vab   = valid + (size_t)b * NPOS;

  float total = 0.f;
  for (int t = wid; t < NPAIR; t += 8) {      // tiles 2t and 2t+1
    int q0 = t * 32 + r16;
    int q1 = q0 + 16;
    const float* Brow0 = Bbase + (size_t)q0 * DDIM + 2*half;
    const float* Brow1 = Brow0 + (size_t)16 * DDIM;
    v8f c0 = {}, c1 = {};
    #pragma unroll 4
    for (int kk = 0; kk < DDIM; kk += 4) {
      v2f a  = *(const v2f*)(Alds  + kk);
      v2f b0 = *(const v2f*)(Brow0 + kk);
      v2f b1 = *(const v2f*)(Brow1 + kk);
      c0 = __builtin_amdgcn_wmma_f32_16x16x4_f32(false, a, false, b0, (short)0, c0, false, false);
      c1 = __builtin_amdgcn_wmma_f32_16x16x4_f32(false, a, false, b1, (short)0, c1, false, false);
    }
    float icol0 = icb[q0], icol1 = icb[q1];
    float vq0   = vab[q0], vq1   = vab[q1];
    float cy0 = (float)(q0 / WC) * 8.f + 4.f;
    float cx0 = (float)(q0 % WC) * 8.f + 4.f;
    float cy1 = (float)(q1 / WC) * 8.f + 4.f;
    float cx1 = (float)(q1 % WC) * 8.f + 4.f;
    float tsum0 = 0.f, tsum1 = 0.f;
    #pragma unroll
    for (int v = 0; v < 8; v++) {
      {
        float g2 = fmaxf(c0[v], 0.f) * irow[v] * icol0;
        float dx = cx0 - wx[v], dy = cy0 - wy[v];
        bool  S  = (dx*dx + dy*dy) <= 56.25f;   // dist <= 7.5
        tsum0 += S ? (250.f * fmaxf(0.f, 1.f - g2)) : fmaxf(0.f, g2 - 0.2f);
      }
      {
        float g2 = fmaxf(c1[v], 0.f) * irow[v] * icol1;
        float dx = cx1 - wx[v], dy = cy1 - wy[v];
        bool  S  = (dx*dx + dy*dy) <= 56.25f;
        tsum1 += S ? (250.f * fmaxf(0.f, 1.f - g2)) : fmaxf(0.f, g2 - 0.2f);
      }
    }
    total += tsum0 * vq0 + tsum1 * vq1;
  }
  #pragma unroll
  for (int m = 16; m >= 1; m >>= 1) total += __shfl_xor(total, m, 32);
  if (lane == 0) atomicAdd(&red, total);
  __syncthreads();
  if (threadIdx.x == 0) atomicAdd(out, red * inorm[b]);
}

// ------------------------------------------------------------------- launcher
extern "C" void kernel_launch(void* const* d_in, const int* in_sizes, int n_in,
                              void* d_out, int out_size, void* d_ws, size_t ws_size,
                              hipStream_t stream) {
  (void)in_sizes; (void)n_in; (void)out_size; (void)ws_size;
  const float* DESC  = (const float*)d_in[0];
  const float* WDESC = (const float*)d_in[1];
  // d_in[2] = H (unused by the reference)
  const float* Hinv  = (const float*)d_in[3];
  const float* vmask = (const float*)d_in[4];
  float* out = (float*)d_out;

  float* ws     = (float*)d_ws;
  float* descN  = ws + OFF_DESCN;
  float* warpN  = ws + OFF_WARPN;
  float* wcoord = ws + OFF_WCOORD;
  float* valid  = ws + OFF_VALID;
  float* rsq    = ws + OFF_RSQ;
  float* invr   = ws + OFF_INVR;
  float* csq    = ws + OFF_CSQ;
  float* invc   = ws + OFF_INVC;
  float* bsum   = ws + OFF_BSUM;
  float* inorm  = ws + OFF_INORM;

  const int nPQ = B_ * NPOS;

  zero_kernel<<<1, 32, 0, stream>>>(out, bsum);
  row_l2norm_kernel<<<2 * nPQ, 256, 0, stream>>>(DESC, WDESC, descN, warpN);
  warp_coords_kernel<<<(nPQ + 255) / 256, 256, 0, stream>>>(Hinv, wcoord);
  valid_kernel<<<(nPQ + 255) / 256, 256, 0, stream>>>(vmask, valid, bsum);
  finalize_norm_kernel<<<1, 32, 0, stream>>>(bsum, inorm);

  // pass 1: row sums of relu(G)^2  -> rsq[p]
  strip_sumsq_kernel<<<B_ * NT, 256, 0, stream>>>(descN, warpN, nullptr, rsq);
  inv_sqrt_kernel<<<(nPQ + 255) / 256, 256, 0, stream>>>(rsq, invr, nPQ);
  // pass 2 (transposed): col sums of (relu(G)*inv_r[p])^2 -> csq[q]
  strip_sumsq_kernel<<<B_ * NT, 256, 0, stream>>>(warpN, descN, invr, csq);
  inv_sqrt_kernel<<<(nPQ + 255) / 256, 256, 0, stream>>>(csq, invc, nPQ);
  // pass 3: fused loss + scalar reduction
  loss_kernel<<<B_ * NT, 256, 0, stream>>>(descN, warpN, invr, invc,
                                           wcoord, valid, inorm, out);
}